// GCN_65274912964781
// MI455X (gfx1250) — compile-verified
//
#include <hip/hip_runtime.h>
#include <hip/hip_bf16.h>

#define IN_DIM 256
#define HID 64

typedef __attribute__((ext_vector_type(2))) float v2f;
typedef __attribute__((ext_vector_type(8))) float v8f;
typedef __attribute__((ext_vector_type(4))) unsigned int v4u;
typedef __attribute__((ext_vector_type(8))) int v8i;
typedef __attribute__((ext_vector_type(4))) int v4i;

// ---------------------------------------------------------------------------
// 1) deg[i] = 1.0f  (self-loop contribution)
// ---------------------------------------------------------------------------
__global__ __launch_bounds__(256) void deg_init_kernel(float* __restrict__ deg, int n) {
    int i = blockIdx.x * 256 + threadIdx.x;
    if (i < n) deg[i] = 1.0f;
}

// ---------------------------------------------------------------------------
// 2) deg[dst[e]] += 1  via HW f32 atomic (global_atomic_add_f32)
// ---------------------------------------------------------------------------
__global__ __launch_bounds__(256) void deg_count_kernel(const long long* __restrict__ dst,
                                                        float* __restrict__ deg,
                                                        long long nEdges) {
    long long e = (long long)blockIdx.x * 256 + threadIdx.x;
    if (e < nEdges) {
        unsafeAtomicAdd(&deg[dst[e]], 1.0f);
    }
}

// ---------------------------------------------------------------------------
// 3) dinv[i] = rsqrt(deg[i])   (deg >= 1 always, matches where(deg>0))
// ---------------------------------------------------------------------------
__global__ __launch_bounds__(256) void deg_rsqrt_kernel(float* __restrict__ deg, int n) {
    int i = blockIdx.x * 256 + threadIdx.x;
    if (i < n) deg[i] = rsqrtf(deg[i]);
}

// ---------------------------------------------------------------------------
// 3.5) Repack W into K-pair-interleaved layout:
//      W2[(p*HID + col)*2 + j] = W[(2p+j)*HID + col],  p = k/2
//      => each WMMA B lane operand {W[kk][col], W[kk+1][col]} is one
//         contiguous 8-byte LDS load into an even-aligned VGPR pair.
// ---------------------------------------------------------------------------
__global__ __launch_bounds__(256) void repack_W_kernel(const float* __restrict__ W,
                                                       float* __restrict__ W2) {
    int i = blockIdx.x * 256 + threadIdx.x;        // [0, 128*64)
    if (i < (IN_DIM / 2) * HID) {
        int p   = i >> 6;
        int col = i & 63;
        W2[i * 2 + 0] = W[(2 * p + 0) * HID + col];
        W2[i * 2 + 1] = W[(2 * p + 1) * HID + col];
    }
}

// ---------------------------------------------------------------------------
// 4) h = X @ W via V_WMMA_F32_16X16X4_F32 (wave32).
//    W2 (64 KB, pair-swizzled) staged into LDS via the Tensor Data Mover
//    (tensor_load_to_lds, TENSORcnt), issued by wave 0 only.
//    Each wave: 16 rows x 64 cols strip, 4 accumulators of v8f.
//
//    A (16x4 f32) lane layout (ISA 7.12.2):
//      lanes 0-15 : M=lane,    VGPR0=K0, VGPR1=K1
//      lanes16-31 : M=lane-16, VGPR0=K2, VGPR1=K3
//    C/D (16x16 f32): VGPR r, lanes 0-15 -> (M=r, N=lane);
//                     lanes 16-31 -> (M=r+8, N=lane-16)
// ---------------------------------------------------------------------------
__global__ __launch_bounds__(256) void gemm_wmma_kernel(const float* __restrict__ x,
                                                        const float* __restrict__ W2,
                                                        float* __restrict__ h,
                                                        int nRows) {
    __shared__ float Ws[IN_DIM * HID];   // 64 KB, pair-swizzled W

    // --- TDM: DMA the whole 64 KB W2 block into LDS (wave 0 issues) ---
    if (threadIdx.x < 32) {
        unsigned long long gaddr = (unsigned long long)(uintptr_t)W2;
        unsigned int lds_off = (unsigned int)(uintptr_t)Ws;   // low 32b = LDS byte offset

        // D# group 0: count=1 | lds_addr | global_addr[56:0] | type=2
        v4u g0;
        g0[0] = 1u;                                           // count=1, user descriptor
        g0[1] = lds_off;
        g0[2] = (unsigned int)(gaddr & 0xFFFFFFFFull);
        g0[3] = (unsigned int)((gaddr >> 32) & 0x1FFFFFFull) | (2u << 30);

        // D# group 1: data_size=4B; tensor = 1 line of 16384 elems; tile 16384x1
        v8i g1;
        g1[0] = (int)(2u << 16);          // data_size=2 (4 bytes); no multicast/pad/iterate
        g1[1] = (int)(16384u << 16);      // tensor_dim0[15:0] @ bits[63:48]
        g1[2] = (int)(1u << 16);          // tensor_dim1[15:0]
        g1[3] = (int)(16384u << 16);      // tile_dim0 @ bits[127:112]
        g1[4] = 1;                        // tile_dim1 = 1
        g1[5] = 16384;                    // tensor_dim0_stride[31:0]
        g1[6] = (int)(16384u << 16);      // tensor_dim1_stride[15:0] @ bits[223:208]
        g1[7] = 0;

        v4i z4 = {0, 0, 0, 0};            // groups 2/3 unused (<=2D tensor)
        v8i z8 = {0, 0, 0, 0, 0, 0, 0, 0};
        __builtin_amdgcn_tensor_load_to_lds(g0, g1, z4, z4, z8, 0);
        __builtin_amdgcn_s_wait_tensorcnt(0);
    }
    __syncthreads();

    const int wave = threadIdx.x >> 5;
    const int lane = threadIdx.x & 31;
    const int tile = blockIdx.x * 8 + wave;   // 16-row tile index (wave-uniform)
    const int row0 = tile * 16;
    if (row0 >= nRows) return;                // wave-uniform: EXEC stays all-1s

    const int half = lane >> 4;               // 0: K+0/K+1, 1: K+2/K+3
    const int l    = lane & 15;               // M (for A) / N (for B)

    int arow = row0 + l;
    if (arow >= nRows) arow = nRows - 1;      // clamp (padding rows never stored)
    const float* __restrict__ xrow = x + (size_t)arow * IN_DIM + half * 2;

    const v2f* __restrict__ Wp = (const v2f*)Ws;

    v8f acc0 = {}, acc1 = {}, acc2 = {}, acc3 = {};

    for (int k = 0; k < IN_DIM; k += 4) {
        v2f a;
        a.x = xrow[k];
        a.y = xrow[k + 1];

        const int pp = (k >> 1) + half;                 // K-pair index
        const v2f* __restrict__ base = Wp + pp * HID + l;
        v2f b0 = base[0];
        v2f b1 = base[16];
        v2f b2 = base[32];
        v2f b3 = base[48];

        acc0 = __builtin_amdgcn_wmma_f32_16x16x4_f32(false, a, false, b0, (short)0, acc0, false, false);
        acc1 = __builtin_amdgcn_wmma_f32_16x16x4_f32(false, a, false, b1, (short)0, acc1, false, false);
        acc2 = __builtin_amdgcn_wmma_f32_16x16x4_f32(false, a, false, b2, (short)0, acc2, false, false);
        acc3 = __builtin_amdgcn_wmma_f32_16x16x4_f32(false, a, false, b3, (short)0, acc3, false, false);
    }

    // Store D tiles
    for (int r = 0; r < 8; ++r) {
        int row = row0 + r + half * 8;
        if (row < nRows) {
            float* __restrict__ hr = h + (size_t)row * HID + l;
            hr[0]  = acc0[r];
            hr[16] = acc1[r];
            hr[32] = acc2[r];
            hr[48] = acc3[r];
        }
    }
}

// ---------------------------------------------------------------------------
// 5) out[i,c] = b[c] + h[i,c] * dinv[i]^2   (self-loop term + bias, inits out)
// ---------------------------------------------------------------------------
__global__ __launch_bounds__(256) void init_out_kernel(const float* __restrict__ h,
                                                       const float* __restrict__ dinv,
                                                       const float* __restrict__ b,
                                                       float* __restrict__ out,
                                                       long long total) {
    long long i = (long long)blockIdx.x * 256 + threadIdx.x;
    if (i < total) {
        long long node = i >> 6;      // /HID
        int c = (int)(i & 63);
        float di = dinv[node];
        out[i] = b[c] + h[i] * di * di;
    }
}

// ---------------------------------------------------------------------------
// 6) scatter: one wave (32 lanes) per edge, 2 columns per lane.
//    out[dst] += h[src] * dinv[src]*dinv[dst]
// ---------------------------------------------------------------------------
__global__ __launch_bounds__(256) void scatter_kernel(const long long* __restrict__ src,
                                                      const long long* __restrict__ dst,
                                                      const float* __restrict__ h,
                                                      const float* __restrict__ dinv,
                                                      float* __restrict__ out,
                                                      long long nEdges) {
    long long e = (long long)blockIdx.x * 8 + (threadIdx.x >> 5);
    if (e >= nEdges) return;
    const int lane = threadIdx.x & 31;

    long long s = src[e];
    long long d = dst[e];
    float w = dinv[s] * dinv[d];

    const float* __restrict__ hs = h + s * HID;
    float* __restrict__ od = out + d * HID;

    unsafeAtomicAdd(&od[lane],      hs[lane]      * w);
    unsafeAtomicAdd(&od[lane + 32], hs[lane + 32] * w);
}

// ---------------------------------------------------------------------------
extern "C" void kernel_launch(void* const* d_in, const int* in_sizes, int n_in,
                              void* d_out, int out_size, void* d_ws, size_t ws_size,
                              hipStream_t stream) {
    const float*      x    = (const float*)d_in[0];       // [N, 256]
    const long long*  eidx = (const long long*)d_in[1];   // [2, E] int64
    const float*      W    = (const float*)d_in[2];       // [256, 64]
    const float*      bias = (const float*)d_in[3];       // [64]
    float*            out  = (float*)d_out;               // [N, 64]

    const int       N = in_sizes[0] / IN_DIM;
    const long long E = in_sizes[1] / 2;

    const long long* srcI = eidx;
    const long long* dstI = eidx + E;

    // workspace layout: deg/dinv [N], h [N*HID], W2 [IN_DIM*HID]
    float* deg = (float*)d_ws;         // becomes dinv in place
    float* h   = deg + N;
    float* W2  = h + (size_t)N * HID;

    // 1) deg = 1
    deg_init_kernel<<<(N + 255) / 256, 256, 0, stream>>>(deg, N);
    // 2) deg += per-edge dst counts
    deg_count_kernel<<<(unsigned)((E + 255) / 256), 256, 0, stream>>>(dstI, deg, E);
    // 3) dinv = rsqrt(deg)
    deg_rsqrt_kernel<<<(N + 255) / 256, 256, 0, stream>>>(deg, N);
    // 3.5) swizzle W for pair-contiguous LDS feeds
    repack_W_kernel<<<((IN_DIM / 2) * HID + 255) / 256, 256, 0, stream>>>(W, W2);
    // 4) h = x @ W (WMMA + TDM-staged weights)
    {
        int tiles  = (N + 15) / 16;
        int blocks = (tiles + 7) / 8;      // 8 waves/block
        gemm_wmma_kernel<<<blocks, 256, 0, stream>>>(x, W2, h, N);
    }
    // 5) out = b + h*dinv^2
    {
        long long total = (long long)N * HID;
        init_out_kernel<<<(unsigned)((total + 255) / 256), 256, 0, stream>>>(h, deg, bias, out, total);
    }
    // 6) scatter-add over edges
    scatter_kernel<<<(unsigned)((E + 7) / 8), 256, 0, stream>>>(srcI, dstI, h, deg, out, E);
}